// SparsificationNetwork_13056700580586
// MI455X (gfx1250) — compile-verified
//
#include <hip/hip_runtime.h>
#include <hip/hip_bf16.h>

// ---------------------------------------------------------------------------
// Sparsification via Gumbel-top-k radix select, MI455X (gfx1250).
//   flows: [32, 2, 512, 512] f32
//   out  : sparse [32,3,512,512] f32  ++  indices [32,4096] i32  ++  mask [32,1,512,512] f32
// Memory-bound: ~192 MiB traffic -> ~8.3us floor @ 23.3 TB/s. Input (64 MiB)
// fits in the 192 MB L2, so the 4 selection sweeps re-read from L2.
// All input streaming goes through the Tensor Data Mover (tensor_load_to_lds,
// double buffered, s_wait_tensorcnt) so loads overlap the log/sqrt VALU work.
// ---------------------------------------------------------------------------

#define BATCH   32
#define HW      (512 * 512)          // 262144 pixels per batch row
#define NSAMP   4096
#define BLOCK   1024                 // 32 wave32 waves
#define CHUNK   4096                 // pixels staged per channel per step (16 KB)
#define NCHUNK  (HW / CHUNK)         // 64
#define BINS    2048                 // 11-bit radix level

typedef __attribute__((ext_vector_type(4))) unsigned int v4u_t;
typedef __attribute__((ext_vector_type(8))) int          v8i_t;
typedef __attribute__((ext_vector_type(4))) int          v4i_t;

#if __has_builtin(__builtin_amdgcn_tensor_load_to_lds) && \
    __has_builtin(__builtin_amdgcn_s_wait_tensorcnt)
#define USE_TDM 1
#else
#define USE_TDM 0
#endif

__device__ __forceinline__ unsigned lds_offset(const void* p) {
  // Generic pointers to LDS carry the byte offset in the low 32 bits.
  return (unsigned)(unsigned long long)p;
}

#if USE_TDM
// 1-D TDM descriptor: count=1, type=image(2), data_size=4B, tile_dim0=nelem,
// tile_dim1=0 (1-D), no padding/iteration/multicast (workgroup_mask must be 0
// outside clusters).
__device__ __forceinline__ void tdm_load_1d(const float* gptr, unsigned lds_off,
                                            int nelem) {
  unsigned long long ga = (unsigned long long)gptr;
  v4u_t g0;
  g0.x = 1u;                                             // count=1, user mode
  g0.y = lds_off;                                        // lds_addr [63:32]
  g0.z = (unsigned)(ga & 0xFFFFFFFFull);                 // global_addr lo
  g0.w = (unsigned)((ga >> 32) & 0x1FFFFFFull)           // global_addr [56:32]
         | (2u << 30);                                   // type = 2 (image)
  v8i_t g1;
  g1[0] = (int)(2u << 16);                               // data_size = 4 bytes
  g1[1] = (int)(((unsigned)nelem & 0xFFFFu) << 16);      // tensor_dim0 lo16
  g1[2] = (int)(((unsigned)nelem >> 16) & 0xFFFFu)       // tensor_dim0 hi16
          | (1 << 16);                                   // tensor_dim1 = 1
  g1[3] = (int)(((unsigned)nelem & 0xFFFFu) << 16);      // tile_dim0
  g1[4] = 0;                                             // tile_dim1/2 unused
  g1[5] = nelem;                                         // dim0 stride (1 row)
  g1[6] = 0;
  g1[7] = 0;
  v4i_t z4 = {0, 0, 0, 0};
#if defined(__clang_major__) && (__clang_major__ >= 23)
  v8i_t z8 = {0, 0, 0, 0, 0, 0, 0, 0};
  __builtin_amdgcn_tensor_load_to_lds(g0, g1, z4, z4, z8, 0);
#else
  __builtin_amdgcn_tensor_load_to_lds(g0, g1, z4, z4, 0);
#endif
}
#endif

// Deterministic surrogate for jax.random.gumbel: per-pixel hash -> Gumbel.
__device__ __forceinline__ unsigned sortable_key(unsigned gid, float fx, float fy) {
  float d  = __fsqrt_rn(fx * fx + fy * fy) * (1.0f / 0.07f);
  float lg = __logf(d + 1e-20f);
  unsigned h = gid * 2654435761u + 0x9E3779B9u;
  h ^= h >> 16; h *= 0x85EBCA6Bu; h ^= h >> 13; h *= 0xC2B2AE35u; h ^= h >> 16;
  float u = (float)(h >> 8) * (1.0f / 16777216.0f) + 7.45e-9f;   // (0,1)
  float g = -__logf(-__logf(u));
  unsigned bits = __float_as_uint(lg + g);
  return bits ^ ((bits & 0x80000000u) ? 0xFFFFFFFFu : 0x80000000u);
}

// Stream both flow channels through double-buffered LDS staging via TDM and
// hand each thread its 4 contiguous pixels per chunk.
template <typename F>
__device__ void sweep_chunks(const float* fx, const float* fy,
                             float (&stage)[2][2][CHUNK], F proc) {
  const int tid = threadIdx.x;
#if USE_TDM
  const bool issuer = (tid < 32);                 // wave 0 only (TDM ignores EXEC)
  if (issuer) {
    tdm_load_1d(fx, lds_offset(&stage[0][0][0]), CHUNK);
    tdm_load_1d(fy, lds_offset(&stage[0][1][0]), CHUNK);
  }
  for (int c = 0; c < NCHUNK; ++c) {
    const int buf = c & 1;
    if (issuer) {
      if (c + 1 < NCHUNK) {
        const int nb = (c + 1) & 1;
        tdm_load_1d(fx + (size_t)(c + 1) * CHUNK, lds_offset(&stage[nb][0][0]), CHUNK);
        tdm_load_1d(fy + (size_t)(c + 1) * CHUNK, lds_offset(&stage[nb][1][0]), CHUNK);
        __builtin_amdgcn_s_wait_tensorcnt(2);     // chunk c's 2 DMAs done (in-order)
      } else {
        __builtin_amdgcn_s_wait_tensorcnt(0);     // drain last chunk
      }
    }
    __syncthreads();                              // staging visible to all waves
    const int p = tid * 4;
    float4 vx = *(const float4*)&stage[buf][0][p];
    float4 vy = *(const float4*)&stage[buf][1][p];
    proc(c * CHUNK + p, vx, vy);
    __syncthreads();                              // done before buffer reuse
  }
#else
  (void)stage;
  for (int c = 0; c < NCHUNK; ++c) {
    const int p = tid * 4;
    float4 vx = *(const float4*)&fx[(size_t)c * CHUNK + p];
    float4 vy = *(const float4*)&fy[(size_t)c * CHUNK + p];
    proc(c * CHUNK + p, vx, vy);
  }
#endif
}

__global__ __launch_bounds__(BLOCK) void sparsify_kernel(
    const float* __restrict__ flows, float* __restrict__ out) {
  __shared__ __align__(16) float stage[2][2][CHUNK];   // 64 KB staging
  __shared__ unsigned hist[BINS];                      // 8 KB radix histogram
  __shared__ unsigned cntA[BLOCK];
  __shared__ unsigned cntT[BLOCK];
  __shared__ unsigned tieTake[BLOCK];
  __shared__ unsigned offSel[BLOCK];
  __shared__ unsigned sT1, sR1, sK22, sR2;

  const int b = blockIdx.x;
  const int tid = threadIdx.x;
  const unsigned gbase = (unsigned)b * HW;
  const float* fx = flows + ((size_t)b * 2 + 0) * HW;
  const float* fy = flows + ((size_t)b * 2 + 1) * HW;

  // ---- Pass 1: histogram of key[31:21] ----
  for (int i = tid; i < BINS; i += BLOCK) hist[i] = 0;
  __syncthreads();
  sweep_chunks(fx, fy, stage, [&](int pix, float4 vx, float4 vy) {
    const float xs[4] = {vx.x, vx.y, vx.z, vx.w};
    const float ys[4] = {vy.x, vy.y, vy.z, vy.w};
#pragma unroll
    for (int j = 0; j < 4; ++j) {
      unsigned k = sortable_key(gbase + (unsigned)(pix + j), xs[j], ys[j]);
      atomicAdd(&hist[k >> 21], 1u);               // ds_add_u32
    }
  });
  __syncthreads();
  if (tid == 0) {
    unsigned acc = 0, R = 1; int T = 0;
    for (int i = BINS - 1; i >= 0; --i) {
      unsigned hv = hist[i];
      if (acc + hv >= NSAMP) { T = i; R = NSAMP - acc; break; }
      acc += hv;
    }
    sT1 = (unsigned)T; sR1 = R;
  }
  __syncthreads();
  const unsigned T1 = sT1;
  const unsigned R1 = sR1;
  __syncthreads();

  // ---- Pass 2: refine key[20:10] inside bin T1 ----
  for (int i = tid; i < BINS; i += BLOCK) hist[i] = 0;
  __syncthreads();
  sweep_chunks(fx, fy, stage, [&](int pix, float4 vx, float4 vy) {
    const float xs[4] = {vx.x, vx.y, vx.z, vx.w};
    const float ys[4] = {vy.x, vy.y, vy.z, vy.w};
#pragma unroll
    for (int j = 0; j < 4; ++j) {
      unsigned k = sortable_key(gbase + (unsigned)(pix + j), xs[j], ys[j]);
      if ((k >> 21) == T1) atomicAdd(&hist[(k >> 10) & 0x7FFu], 1u);
    }
  });
  __syncthreads();
  if (tid == 0) {
    unsigned acc = 0, R = 1; int T = 0;
    for (int i = BINS - 1; i >= 0; --i) {
      unsigned hv = hist[i];
      if (acc + hv >= R1) { T = i; R = R1 - acc; break; }
      acc += hv;
    }
    sK22 = (T1 << 11) | (unsigned)T;
    sR2 = R;
  }
  __syncthreads();
  const unsigned K22 = sK22;
  const unsigned R2 = sR2;

  // ---- Pass 3: per-thread counts -> deterministic offsets & tie quotas ----
  unsigned nA = 0, nT = 0;
  sweep_chunks(fx, fy, stage, [&](int pix, float4 vx, float4 vy) {
    const float xs[4] = {vx.x, vx.y, vx.z, vx.w};
    const float ys[4] = {vy.x, vy.y, vy.z, vy.w};
#pragma unroll
    for (int j = 0; j < 4; ++j) {
      unsigned kh = sortable_key(gbase + (unsigned)(pix + j), xs[j], ys[j]) >> 10;
      nA += (kh > K22);
      nT += (kh == K22);
    }
  });
  cntA[tid] = nA; cntT[tid] = nT;
  __syncthreads();
  if (tid == 0) {                                  // exactly NSAMP selected total
    unsigned accT = 0, accS = 0;
    for (int t = 0; t < BLOCK; ++t) {
      unsigned ta = cntA[t], tt = cntT[t], take = 0;
      if (accT < R2) { take = R2 - accT; if (take > tt) take = tt; }
      tieTake[t] = take;
      offSel[t] = accS;
      accS += ta + take;
      accT += tt;
    }
  }
  __syncthreads();
  const unsigned myTake = tieTake[tid];
  unsigned writeAt = offSel[tid];

  // ---- Pass 4: emit sparse flows, mask, mask channel, indices ----
  float* oFx   = out + ((size_t)b * 3 + 0) * HW;
  float* oFy   = out + ((size_t)b * 3 + 1) * HW;
  float* oM    = out + ((size_t)b * 3 + 2) * HW;
  int*   oIdx  = (int*)(out + (size_t)BATCH * 3 * HW) + (size_t)b * NSAMP;
  float* oMask = out + (size_t)BATCH * 3 * HW + (size_t)BATCH * NSAMP + (size_t)b * HW;
  unsigned tieSeen = 0;
  sweep_chunks(fx, fy, stage, [&](int pix, float4 vx, float4 vy) {
    const float xs[4] = {vx.x, vx.y, vx.z, vx.w};
    const float ys[4] = {vy.x, vy.y, vy.z, vy.w};
    float mx[4], my[4], mm[4];
#pragma unroll
    for (int j = 0; j < 4; ++j) {
      unsigned kh = sortable_key(gbase + (unsigned)(pix + j), xs[j], ys[j]) >> 10;
      const bool tie = (kh == K22);
      const bool sel = (kh > K22) || (tie && (tieSeen < myTake));
      tieSeen += tie ? 1u : 0u;
      const float m = sel ? 1.0f : 0.0f;
      mx[j] = xs[j] * m; my[j] = ys[j] * m; mm[j] = m;
      if (sel) { oIdx[writeAt] = pix + j; ++writeAt; }
    }
    *(float4*)&oFx[pix]   = make_float4(mx[0], mx[1], mx[2], mx[3]);
    *(float4*)&oFy[pix]   = make_float4(my[0], my[1], my[2], my[3]);
    *(float4*)&oM[pix]    = make_float4(mm[0], mm[1], mm[2], mm[3]);
    *(float4*)&oMask[pix] = make_float4(mm[0], mm[1], mm[2], mm[3]);
  });
}

extern "C" void kernel_launch(void* const* d_in, const int* in_sizes, int n_in,
                              void* d_out, int out_size, void* d_ws, size_t ws_size,
                              hipStream_t stream) {
  (void)in_sizes; (void)n_in; (void)d_ws; (void)ws_size; (void)out_size;
  const float* flows = (const float*)d_in[0];
  float* out = (float*)d_out;
  hipLaunchKernelGGL(sparsify_kernel, dim3(BATCH), dim3(BLOCK), 0, stream,
                     flows, out);
}